// MemoryLayer_65060164600207
// MI455X (gfx1250) — compile-verified
//
#include <hip/hip_runtime.h>
#include <hip/hip_bf16.h>
#include <math.h>

typedef float v2f __attribute__((ext_vector_type(2)));
typedef float v8f __attribute__((ext_vector_type(8)));

// D = A(16x4) * B(4x16) + C, fp32 WMMA (CDNA5 V_WMMA_F32_16X16X4_F32)
__device__ __forceinline__ v8f wmma_f32(v2f a, v2f b, v8f c) {
    return __builtin_amdgcn_wmma_f32_16x16x4_f32(false, a, false, b, (short)0, c,
                                                 false, false);
}

__device__ __forceinline__ float wave_sum(float v) {
    for (int off = 16; off > 0; off >>= 1) v += __shfl_xor(v, off, 32);
    return v;
}

// ---------------- constants ----------------
#define Bn   4
#define Sn   128
#define Cn   49
#define Mn   288
#define BERT 768
#define CP1  50          // C+1
#define H1   50
#define W1   288
#define H2   25
#define W2   144

// workspace offsets (floats)
#define OFF_EB     0u            // 512*288
#define OFF_ES     147456u       // 512*288  (== t1)
#define OFF_T2     294912u       // 512*288
#define OFF_V      442368u       // 512*288
#define OFF_EBSUM  589824u       // 4*288
#define OFF_W1S    590976u       // 288
#define OFF_W2S    591264u       // 288
#define OFF_SCORE  591552u       // 512*50
#define OFF_ATT    617152u       // 512*50
#define OFF_AW     642752u       // 4*128*128
#define OFF_XM     708288u       // 4*128*144
#define OFF_SUMM   782016u       // 4*128*50*288
#define OFF_C1     8154816u      // 4*128*50*288
#define OFF_POOL   15527616u     // 4*128*25*144
#define OFF_C2     17370816u     // 4*128*25*144

// ---------------- K1: eb = bert @ W_lin1^T + b ----------------
// rows = B*S = 512 (M), cols = 288 (N), K = 768
__global__ void eb_gemm(const float* __restrict__ bert, const float* __restrict__ Wl,
                        const float* __restrict__ bl, float* __restrict__ eb) {
    int lane = threadIdx.x & 31, wave = threadIdx.x >> 5;
    int tile = blockIdx.x * 8 + wave;
    if (tile >= 32 * 18) return;                 // wave-uniform
    int m0 = (tile / 18) * 16, n0 = (tile % 18) * 16;
    int half = lane >> 4, l16 = lane & 15;
    v8f acc = {};
    for (int k0 = 0; k0 < BERT; k0 += 4) {
        int k = k0 + 2 * half;
        v2f a, b;
        a.x = bert[(m0 + l16) * BERT + k];
        a.y = bert[(m0 + l16) * BERT + k + 1];
        b.x = Wl[(n0 + l16) * BERT + k];         // B[K=k][N=m] = W_lin1[m][k]
        b.y = Wl[(n0 + l16) * BERT + k + 1];
        acc = wmma_f32(a, b, acc);
    }
    for (int r = 0; r < 8; ++r) {
        int row = m0 + r + 8 * half, col = n0 + l16;
        eb[row * Mn + col] = acc[r] + bl[col];
    }
}

// ---------------- K1b: EBsum[b,m] = sum_t eb ----------------
__global__ void ebsum_k(const float* __restrict__ eb, float* __restrict__ ebsum) {
    int i = blockIdx.x * blockDim.x + threadIdx.x;
    if (i >= Bn * Mn) return;
    int b = i / Mn, m = i % Mn;
    float s = 0.f;
    for (int t = 0; t < Sn; ++t) s += eb[(b * Sn + t) * Mn + m];
    ebsum[i] = s;
}

// ---------------- K0: weight row-sums ----------------
__global__ void wsums_k(const float* __restrict__ w1, const float* __restrict__ w2,
                        float* __restrict__ w1s, float* __restrict__ w2s) {
    int p = blockIdx.x * blockDim.x + threadIdx.x;
    if (p >= Mn) return;
    float a = 0.f, b = 0.f;
    for (int c = 0; c < CP1; ++c) { a += w1[p * CP1 + c]; b += w2[p * CP1 + c]; }
    w1s[p] = a; w2s[p] = b;
}

// ---------------- K2: ES + score per (b,s) ----------------
__global__ void es_score_k(const float* __restrict__ mem, const float* __restrict__ sent,
                           const float* __restrict__ ebsum, float* __restrict__ ES,
                           float* __restrict__ score) {
    int bs = blockIdx.x;                         // b*128 + s
    __shared__ float ebs[Mn];
    for (int i = threadIdx.x; i < Mn; i += 256) ebs[i] = ebsum[(bs >> 7) * Mn + i];
    __syncthreads();
    const float* mrow = mem + (size_t)bs * Cn * Mn;
    for (int m = threadIdx.x; m < Mn; m += 256) {   // column sums (+sentinel)
        float s = sent[m];
        for (int c = 0; c < Cn; ++c) s += mrow[c * Mn + m];
        ES[(size_t)bs * Mn + m] = s;
    }
    int wave = threadIdx.x >> 5, lane = threadIdx.x & 31;
    for (int j = wave; j < CP1; j += 8) {           // row dots with EBsum
        const float* row = (j == 0) ? sent : (mrow + (j - 1) * Mn);
        float p = 0.f;
        for (int m = lane; m < Mn; m += 32) p += row[m] * ebs[m];
        p = wave_sum(p);
        if (lane == 0) score[bs * CP1 + j] = p * (1.0f / (float)Sn);
    }
}

// ---------------- K3: attention (mask, blended softmax, top-5, softmax) ----------------
__global__ void attention_k(const float* __restrict__ score, const int* __restrict__ mlen,
                            const float* __restrict__ cnw, float* __restrict__ att) {
    int bs = blockIdx.x;
    int j = threadIdx.x;                         // blockDim = 64
    __shared__ float sv[CP1], tmp[CP1], s2[CP1];
    float ml = (float)mlen[bs];
    if (j < CP1) sv[j] = score[bs * CP1 + j] + ((ml < (float)j) ? -1e6f : 0.f);
    __syncthreads();
    float p1 = 0.f, p2 = 0.f;
    if (j < CP1) {
        float mx = -1e30f;
        for (int k = 0; k < CP1; ++k) mx = fmaxf(mx, sv[k]);
        tmp[j] = expf(sv[j] - mx);
    }
    __syncthreads();
    if (j < CP1) {
        float sm = 0.f;
        for (int k = 0; k < CP1; ++k) sm += tmp[k];
        p1 = tmp[j] / sm;
    }
    __syncthreads();
    if (j < CP1) sv[j] = (j == 0) ? 0.f : cnw[bs * Cn + (j - 1)];
    __syncthreads();
    if (j < CP1) {
        float mx = -1e30f;
        for (int k = 0; k < CP1; ++k) mx = fmaxf(mx, sv[k]);
        tmp[j] = expf(sv[j] - mx);
    }
    __syncthreads();
    if (j < CP1) {
        float sm = 0.f;
        for (int k = 0; k < CP1; ++k) sm += tmp[k];
        p2 = tmp[j] / sm;
        s2[j] = 0.7f * p1 + 0.3f * p2;
    }
    __syncthreads();
    if (j < CP1) {                               // keep iff < 5 strictly-greater elems
        int cnt = 0;
        for (int k = 0; k < CP1; ++k) cnt += (s2[k] > s2[j]) ? 1 : 0;
        sv[j] = (cnt < 5) ? s2[j] : 0.f;
    }
    __syncthreads();
    if (j < CP1) {
        float mx = -1e30f;
        for (int k = 0; k < CP1; ++k) mx = fmaxf(mx, sv[k]);
        tmp[j] = expf(sv[j] - mx);
    }
    __syncthreads();
    if (j < CP1) {
        float sm = 0.f;
        for (int k = 0; k < CP1; ++k) sm += tmp[k];
        att[bs * CP1 + j] = tmp[j] / sm;
    }
}

// ---------------- K4: AWsum[b] = ES[b] @ eb[b]^T  ([128s x 128t], K=288) ----------------
__global__ void aw_gemm(const float* __restrict__ ES, const float* __restrict__ eb,
                        float* __restrict__ AW) {
    int lane = threadIdx.x & 31, wave = threadIdx.x >> 5;
    int tile = blockIdx.x * 8 + wave;            // 4*8*8 = 256 tiles
    int b = tile >> 6, rem = tile & 63;
    int m0 = (rem >> 3) * 16, n0 = (rem & 7) * 16;
    int half = lane >> 4, l16 = lane & 15;
    v8f acc = {};
    for (int k0 = 0; k0 < Mn; k0 += 4) {
        int k = k0 + 2 * half;
        v2f a, bb;
        a.x  = ES[((size_t)(b * Sn + m0 + l16)) * Mn + k];
        a.y  = ES[((size_t)(b * Sn + m0 + l16)) * Mn + k + 1];
        bb.x = eb[((size_t)(b * Sn + n0 + l16)) * Mn + k];   // B[K=m][N=t] = eb[t][m]
        bb.y = eb[((size_t)(b * Sn + n0 + l16)) * Mn + k + 1];
        acc = wmma_f32(a, bb, acc);
    }
    for (int r = 0; r < 8; ++r)
        AW[(size_t)(b * Sn + m0 + r + 8 * half) * Sn + n0 + l16] = acc[r];
}

// ---------------- K5: t2[b] = AW[b] @ eb[b]  ([128s x 288m], K=128) ----------------
__global__ void t2_gemm(const float* __restrict__ AW, const float* __restrict__ eb,
                        float* __restrict__ t2) {
    int lane = threadIdx.x & 31, wave = threadIdx.x >> 5;
    int tile = blockIdx.x * 8 + wave;            // 4*8*18 = 576 tiles
    int b = tile / 144, rem = tile % 144;
    int m0 = (rem / 18) * 16, n0 = (rem % 18) * 16;
    int half = lane >> 4, l16 = lane & 15;
    v8f acc = {};
    for (int k0 = 0; k0 < Sn; k0 += 4) {
        int k = k0 + 2 * half;
        v2f a, bb;
        a.x  = AW[(size_t)(b * Sn + m0 + l16) * Sn + k];
        a.y  = AW[(size_t)(b * Sn + m0 + l16) * Sn + k + 1];
        bb.x = eb[(size_t)(b * Sn + k) * Mn + n0 + l16];     // B[K=t][N=m] = eb[t][m]
        bb.y = eb[(size_t)(b * Sn + k + 1) * Mn + n0 + l16];
        acc = wmma_f32(a, bb, acc);
    }
    for (int r = 0; r < 8; ++r)
        t2[(size_t)(b * Sn + m0 + r + 8 * half) * Mn + n0 + l16] = acc[r];
}

// ---------------- K6: v[b,s,n] = softmax-weighted sum over p ----------------
__global__ void cwt_v_k(const float* __restrict__ ES, const float* __restrict__ t2,
                        const float* __restrict__ w1s, const float* __restrict__ w2s,
                        float* __restrict__ v) {
    int bs = blockIdx.x;
    int n = threadIdx.x;                          // blockDim = 288
    __shared__ float w1l[Mn], w2l[Mn], t1l[Mn];
    w1l[n] = w1s[n]; w2l[n] = w2s[n]; t1l[n] = ES[(size_t)bs * Mn + n];
    __syncthreads();
    float x = t1l[n], y = t2[(size_t)bs * Mn + n];
    float num = 0.f, den = 0.f;
    for (int p = 0; p < Mn; ++p) {
        float e = expf(tanhf(w1l[p] * x + w2l[p] * y));
        num += t1l[p] * e;
        den += e;
    }
    v[(size_t)bs * Mn + n] = num / den;
}

// ---------------- K7: summ[b,s,a,n] = att * v (conv input, NCHW) ----------------
__global__ void build_summ(const float* __restrict__ att, const float* __restrict__ v,
                           float* __restrict__ summ) {
    size_t idx = (size_t)blockIdx.x * blockDim.x + threadIdx.x;  // 7,372,800
    int n = (int)(idx % Mn);
    size_t r = idx / Mn;
    int a = (int)(r % CP1);
    int bs = (int)(r / CP1);
    summ[idx] = att[bs * CP1 + a] * v[(size_t)bs * Mn + n];
}

// ---------------- conv 5x5 same-pad, Cin=Cout=128, + bias + ReLU (WMMA) ----------------
// M = co (16 per wave), N = w (16 per block), K = ci x taps
__global__ void conv5x5_wmma(const float* __restrict__ in, const float* __restrict__ wgt,
                             const float* __restrict__ bias, float* __restrict__ out,
                             int H, int W) {
    int lane = threadIdx.x & 31, wave = threadIdx.x >> 5;
    int half = lane >> 4, l16 = lane & 15;
    int co0 = wave * 16;
    int w0 = blockIdx.x * 16;
    int h = blockIdx.y;
    int n = blockIdx.z;
    size_t HW = (size_t)H * W;
    v8f acc = {};
    for (int dy = 0; dy < 5; ++dy) {
        int ih = h + dy - 2;
        if (ih < 0 || ih >= H) continue;
        for (int dx = 0; dx < 5; ++dx) {
            int iw = w0 + l16 + dx - 2;
            bool inb = (iw >= 0) && (iw < W);
            const float* ip = in + (size_t)(n * 128) * HW + (size_t)ih * W + iw;
            const float* wp = wgt + dy * 5 + dx;
            for (int k0 = 0; k0 < 128; k0 += 4) {
                int ci = k0 + 2 * half;
                v2f a, b;
                a.x = wp[((co0 + l16) * 128 + ci) * 25];
                a.y = wp[((co0 + l16) * 128 + ci + 1) * 25];
                b.x = inb ? ip[(size_t)ci * HW] : 0.0f;
                b.y = inb ? ip[(size_t)(ci + 1) * HW] : 0.0f;
                acc = wmma_f32(a, b, acc);
            }
        }
    }
    for (int r = 0; r < 8; ++r) {
        int co = co0 + r + 8 * half;
        float val = acc[r] + bias[co];
        out[(size_t)(n * 128 + co) * HW + (size_t)h * W + (w0 + l16)] = fmaxf(val, 0.0f);
    }
}

// ---------------- maxpool 2x2 ----------------
__global__ void maxpool_k(const float* __restrict__ in, float* __restrict__ out) {
    int idx = blockIdx.x * blockDim.x + threadIdx.x;      // 1,843,200
    int wo = idx % W2;
    int t = idx / W2;
    int ho = t % H2;
    int nc = t / H2;
    const float* p = in + ((size_t)nc * H1 + 2 * ho) * W1 + 2 * wo;
    out[idx] = fmaxf(fmaxf(p[0], p[1]), fmaxf(p[W1], p[W1 + 1]));
}

// ---------------- LayerNorm(144) in place, one wave per row ----------------
__global__ void layernorm_k(float* __restrict__ x, const float* __restrict__ g,
                            const float* __restrict__ b) {
    int wave = threadIdx.x >> 5, lane = threadIdx.x & 31;
    int row = blockIdx.x * 8 + wave;                      // 12800 rows
    if (row >= Bn * Sn * H2) return;
    float* rp = x + (size_t)row * W2;
    float vals[5];
    float s = 0.f;
#pragma unroll
    for (int q = 0; q < 5; ++q) {
        int f = lane + 32 * q;
        vals[q] = (f < W2) ? rp[f] : 0.f;
        s += vals[q];
    }
    s = wave_sum(s);
    float mu = s / (float)W2;
    float vs = 0.f;
#pragma unroll
    for (int q = 0; q < 5; ++q) {
        int f = lane + 32 * q;
        if (f < W2) { float d = vals[q] - mu; vs += d * d; }
    }
    vs = wave_sum(vs);
    float rstd = rsqrtf(vs / (float)W2 + 1e-5f);
#pragma unroll
    for (int q = 0; q < 5; ++q) {
        int f = lane + 32 * q;
        if (f < W2) rp[f] = (vals[q] - mu) * rstd * g[f] + b[f];
    }
}

// ---------------- mean over H2 ----------------
__global__ void hmean_k(const float* __restrict__ x, float* __restrict__ xm) {
    int idx = blockIdx.x * blockDim.x + threadIdx.x;      // 73728
    if (idx >= Bn * Sn * W2) return;
    int f = idx % W2;
    int nc = idx / W2;
    float s = 0.f;
    for (int h = 0; h < H2; ++h) s += x[((size_t)nc * H2 + h) * W2 + f];
    xm[idx] = s * (1.0f / (float)H2);
}

// ---------------- final linear 144 -> 300 + mask ----------------
__global__ void final_k(const float* __restrict__ xm, const float* __restrict__ W,
                        const float* __restrict__ bo, const int* __restrict__ mlen,
                        const int* __restrict__ ign, float* __restrict__ out) {
    int idx = blockIdx.x * blockDim.x + threadIdx.x;      // 153600
    if (idx >= Bn * Sn * 300) return;
    int o = idx % 300;
    int bs = idx / 300;
    float s = bo[o];
    for (int f = 0; f < W2; ++f) s += xm[(size_t)bs * W2 + f] * W[o * W2 + f];
    if (*ign && mlen[bs] <= 0) s = 0.f;
    out[idx] = s;
}

extern "C" void kernel_launch(void* const* d_in, const int* in_sizes, int n_in,
                              void* d_out, int out_size, void* d_ws, size_t ws_size,
                              hipStream_t stream) {
    const float* bert = (const float*)d_in[0];
    const float* mem  = (const float*)d_in[1];
    const int*   mlen = (const int*)d_in[2];
    const float* cnw  = (const float*)d_in[3];
    const int*   ign  = (const int*)d_in[4];
    const float* Wl1  = (const float*)d_in[6];
    const float* bl1  = (const float*)d_in[7];
    const float* sent = (const float*)d_in[8];
    const float* w1   = (const float*)d_in[9];
    const float* w2   = (const float*)d_in[10];
    const float* c1w  = (const float*)d_in[11];
    const float* c1b  = (const float*)d_in[12];
    const float* c2w  = (const float*)d_in[13];
    const float* c2b  = (const float*)d_in[14];
    const float* lng  = (const float*)d_in[15];
    const float* lnb  = (const float*)d_in[16];
    const float* Wout = (const float*)d_in[17];
    const float* bout = (const float*)d_in[18];

    float* ws = (float*)d_ws;
    float* eb    = ws + OFF_EB;
    float* ES    = ws + OFF_ES;
    float* t2    = ws + OFF_T2;
    float* v     = ws + OFF_V;
    float* ebsum = ws + OFF_EBSUM;
    float* w1s   = ws + OFF_W1S;
    float* w2s   = ws + OFF_W2S;
    float* score = ws + OFF_SCORE;
    float* att   = ws + OFF_ATT;
    float* AW    = ws + OFF_AW;
    float* xm    = ws + OFF_XM;
    float* summ  = ws + OFF_SUMM;
    float* c1o   = ws + OFF_C1;
    float* pool  = ws + OFF_POOL;
    float* c2o   = ws + OFF_C2;

    eb_gemm<<<72, 256, 0, stream>>>(bert, Wl1, bl1, eb);
    ebsum_k<<<5, 256, 0, stream>>>(eb, ebsum);
    wsums_k<<<2, 256, 0, stream>>>(w1, w2, w1s, w2s);
    es_score_k<<<512, 256, 0, stream>>>(mem, sent, ebsum, ES, score);
    attention_k<<<512, 64, 0, stream>>>(score, mlen, cnw, att);
    aw_gemm<<<32, 256, 0, stream>>>(ES, eb, AW);
    t2_gemm<<<72, 256, 0, stream>>>(AW, eb, t2);
    cwt_v_k<<<512, 288, 0, stream>>>(ES, t2, w1s, w2s, v);
    build_summ<<<28800, 256, 0, stream>>>(att, v, summ);
    conv5x5_wmma<<<dim3(W1 / 16, H1, Bn), 256, 0, stream>>>(summ, c1w, c1b, c1o, H1, W1);
    maxpool_k<<<7200, 256, 0, stream>>>(c1o, pool);
    conv5x5_wmma<<<dim3(W2 / 16, H2, Bn), 256, 0, stream>>>(pool, c2w, c2b, c2o, H2, W2);
    layernorm_k<<<1600, 256, 0, stream>>>(c2o, lng, lnb);
    hmean_k<<<288, 256, 0, stream>>>(c2o, xm);
    final_k<<<600, 256, 0, stream>>>(xm, Wout, bout, mlen, ign, (float*)d_out);
}